// SimpleGNN_2147483648472
// MI455X (gfx1250) — compile-verified
//
#include <hip/hip_runtime.h>

#define N_NODES  500000
#define N_EDGES  5000000
#define N_GRAPHS 5000
#define FEAT     7
#define STATE    16
#define ROUNDS   4
#define NTILES   (N_NODES / 16)          // 31250 exact
#define CHUNK    32
#define NCHUNK   (N_NODES / CHUNK)       // 15625 exact

typedef __attribute__((ext_vector_type(2))) float v2f;
typedef __attribute__((ext_vector_type(8))) float v8f;

// One wave computes D[16x16] = A[16x16] @ W[16x16] using 4 chained
// V_WMMA_F32_16X16X4_F32 ops (K split into 4 chunks of 4).
// A layout (ISA 7.12.2, 32-bit A 16x4): lanes 0-15 hold K=4kc,4kc+1 of row
// m=lane; lanes 16-31 hold K=4kc+2,4kc+3. B layout (4x16): K halves split
// across lane halves (VGPR0: K=kb / VGPR1: K=kb+1, kb = 4kc + 2*(lane>>4)).
// Both reduce to: a = {A[m][kb], A[m][kb+1]}, b = {W[kb][n], W[kb+1][n]}
// with m = n = lane&15.
__device__ __forceinline__ v8f wmma_tile16(const float* __restrict__ A,
                                           const float* __restrict__ W) {
  const int lane = threadIdx.x & 31;
  const int mn   = lane & 15;
  const int kh   = (lane >> 4) << 1;   // 0 or 2
  v8f c = {0.f, 0.f, 0.f, 0.f, 0.f, 0.f, 0.f, 0.f};
#pragma unroll
  for (int kc = 0; kc < 4; ++kc) {
    const int kb = 4 * kc + kh;
    v2f a, b;
    a.x = A[mn * 16 + kb];
    a.y = A[mn * 16 + kb + 1];
    b.x = W[kb * 16 + mn];
    b.y = W[(kb + 1) * 16 + mn];
    c = __builtin_amdgcn_wmma_f32_16x16x4_f32(false, a, false, b, (short)0, c,
                                              false, false);
  }
  return c;
}

// Fused: state = relu(x @ in_W + in_b); message = relu(state @ msg_W0 + b0);
// zero agg tile; zero graph accumulator. K=7 padded to 8 via clamped index +
// 0/1 mask (branch-free, EXEC stays all-ones for the WMMAs).
__global__ void k_input_msg(const float* __restrict__ x, const float* __restrict__ inW,
                            const float* __restrict__ inb,
                            const float* __restrict__ msgW0, const float* __restrict__ msgb0,
                            float* __restrict__ state, float* __restrict__ message,
                            float* __restrict__ agg, float* __restrict__ gacc) {
  const int gid = blockIdx.x * blockDim.x + threadIdx.x;
  if (gid < N_GRAPHS * STATE) gacc[gid] = 0.f;
  const int wave = gid >> 5;
  if (wave >= NTILES) return;
  const int lane = threadIdx.x & 31;
  const int mn   = lane & 15;
  const int kh   = (lane >> 4) << 1;
  const float* xrow = x + (size_t)wave * 16 * FEAT;
  v8f c = {0.f, 0.f, 0.f, 0.f, 0.f, 0.f, 0.f, 0.f};
#pragma unroll
  for (int kc = 0; kc < 2; ++kc) {
    const int kb = 4 * kc + kh;
    const float m0 = (kb     < FEAT) ? 1.f : 0.f;
    const float m1 = (kb + 1 < FEAT) ? 1.f : 0.f;
    int ia0 = mn * FEAT + kb;       ia0 = ia0 < 16 * FEAT     ? ia0 : 0;
    int ia1 = mn * FEAT + kb + 1;   ia1 = ia1 < 16 * FEAT     ? ia1 : 0;
    int ib0 = kb * STATE + mn;      ib0 = ib0 < FEAT * STATE  ? ib0 : 0;
    int ib1 = (kb + 1) * STATE + mn; ib1 = ib1 < FEAT * STATE ? ib1 : 0;
    v2f a, b;
    a.x = xrow[ia0] * m0;
    a.y = xrow[ia1] * m1;
    b.x = inW[ib0] * m0;
    b.y = inW[ib1] * m1;
    c = __builtin_amdgcn_wmma_f32_16x16x4_f32(false, a, false, b, (short)0, c,
                                              false, false);
  }
  const float bn = inb[mn];
  const int   mh = (lane >> 4) * 8;
  float* srow = state + (size_t)wave * 256;
#pragma unroll
  for (int i = 0; i < 8; ++i) {
    float v = c[i] + bn;
    srow[(mh + i) * 16 + mn] = v > 0.f ? v : 0.f;
  }
  // Round-0 message from the freshly written tile. Same-wave store->load to
  // the same addresses is kept in order (ISA VMEM ordering); hits L0.
  v8f m = wmma_tile16(srow, msgW0);
  const float mb = msgb0[mn];
  float* mrow = message + (size_t)wave * 256;
  float* arow = agg + (size_t)wave * 256;
#pragma unroll
  for (int i = 0; i < 8; ++i) {
    float v = m[i] + mb;
    mrow[(mh + i) * 16 + mn] = v > 0.f ? v : 0.f;
    arow[(mh + i) * 16 + mn] = 0.f;
  }
}

// One thread per (edge, feature): agg[dst] += message[src]. 16 consecutive
// lanes hit 64 contiguous bytes -> coalesced f32 atomics resolved in L2
// (entire working set is L2-resident on the 192 MB MI455X L2).
__global__ void k_scatter(const float* __restrict__ message,
                          const int* __restrict__ ei, float* __restrict__ agg) {
  const unsigned t = blockIdx.x * blockDim.x + threadIdx.x;
  if (t >= (unsigned)N_EDGES * 16u) return;
  const int e = t >> 4;
  const int f = t & 15;
  const int s = ei[e];             // edge_index[0] = src
  const int d = ei[N_EDGES + e];   // edge_index[1] = dst
  atomicAdd(&agg[(size_t)d * 16 + f], message[(size_t)s * 16 + f]);
}

// Fused: state += relu(agg @ upd_W[r] + b); message = relu(state @ msg_W[r+1] + b);
// re-zero agg tile for the next scatter.
__global__ void k_upd_msg(float* __restrict__ agg,
                          const float* __restrict__ updW, const float* __restrict__ updb,
                          const float* __restrict__ msgWn, const float* __restrict__ msgbn,
                          float* __restrict__ state, float* __restrict__ message) {
  const int wave = (blockIdx.x * blockDim.x + threadIdx.x) >> 5;
  if (wave >= NTILES) return;
  const int lane = threadIdx.x & 31;
  float* arow = agg + (size_t)wave * 256;
  v8f c = wmma_tile16(arow, updW);
  // All agg loads have retired (s_wait_loadcnt precedes the consuming WMMA);
  // the clobber keeps the zero-stores below from being scheduled earlier.
  asm volatile("" ::: "memory");
  const int   n  = lane & 15;
  const int   mh = (lane >> 4) * 8;
  const float bn = updb[n];
  float* srow = state + (size_t)wave * 256;
#pragma unroll
  for (int i = 0; i < 8; ++i) {
    float v = c[i] + bn;
    v = v > 0.f ? v : 0.f;
    srow[(mh + i) * 16 + n] += v;
    arow[(mh + i) * 16 + n] = 0.f;
  }
  // Next-round message from updated state (same-wave store->load ordered).
  v8f m = wmma_tile16(srow, msgWn);
  const float mb = msgbn[n];
  float* mrow = message + (size_t)wave * 256;
#pragma unroll
  for (int i = 0; i < 8; ++i) {
    float v = m[i] + mb;
    mrow[(mh + i) * 16 + n] = v > 0.f ? v : 0.f;
  }
}

// Final round: state += relu(agg @ W + b) only.
__global__ void k_upd(const float* __restrict__ agg, const float* __restrict__ W,
                      const float* __restrict__ bias, float* __restrict__ state) {
  const int wave = (blockIdx.x * blockDim.x + threadIdx.x) >> 5;
  if (wave >= NTILES) return;
  const int lane = threadIdx.x & 31;
  v8f c = wmma_tile16(agg + (size_t)wave * 256, W);
  const int   n  = lane & 15;
  const int   mh = (lane >> 4) * 8;
  const float bn = bias[n];
  float* srow = state + (size_t)wave * 256;
#pragma unroll
  for (int i = 0; i < 8; ++i) {
    float v = c[i] + bn;
    v = v > 0.f ? v : 0.f;
    srow[(mh + i) * 16 + n] += v;
  }
}

// Segment-sum over sorted batch: each thread owns a 32-node contiguous chunk
// for one feature; run-length accumulate, flush one atomic per graph-id run.
__global__ void k_gsum(const float* __restrict__ state, const int* __restrict__ batch,
                       float* __restrict__ gacc) {
  const int t = blockIdx.x * blockDim.x + threadIdx.x;
  if (t >= NCHUNK * 16) return;
  const int f  = t & 15;
  const int ch = t >> 4;
  const int n0 = ch * CHUNK;
  const int n1 = n0 + CHUNK;
  int   cur = batch[n0];
  float acc = 0.f;
  for (int n = n0; n < n1; ++n) {
    const int bn = batch[n];
    if (bn != cur) {
      atomicAdd(&gacc[(size_t)cur * 16 + f], acc);
      acc = 0.f;
      cur = bn;
    }
    acc += state[(size_t)n * 16 + f];
  }
  atomicAdd(&gacc[(size_t)cur * 16 + f], acc);
}

// out[g] = graph_state[g] . out_W + out_b
__global__ void k_out(const float* __restrict__ gacc, const float* __restrict__ outW,
                      const float* __restrict__ outb, float* __restrict__ out) {
  const int g = blockIdx.x * blockDim.x + threadIdx.x;
  if (g >= N_GRAPHS) return;
  float s = outb[0];
#pragma unroll
  for (int k = 0; k < STATE; ++k) s += gacc[(size_t)g * 16 + k] * outW[k];
  out[g] = s;
}

extern "C" void kernel_launch(void* const* d_in, const int* in_sizes, int n_in,
                              void* d_out, int out_size, void* d_ws, size_t ws_size,
                              hipStream_t stream) {
  const float* x     = (const float*)d_in[0];
  const int*   ei    = (const int*)d_in[1];
  const int*   batch = (const int*)d_in[2];
  const float* inW   = (const float*)d_in[3];
  const float* inb   = (const float*)d_in[4];
  const float* msgW  = (const float*)d_in[5];
  const float* msgb  = (const float*)d_in[6];
  const float* updW  = (const float*)d_in[7];
  const float* updb  = (const float*)d_in[8];
  const float* outW  = (const float*)d_in[9];
  const float* outb  = (const float*)d_in[10];
  float* out = (float*)d_out;

  float* state   = (float*)d_ws;                         // 32 MB
  float* message = state + (size_t)N_NODES * STATE;      // 32 MB
  float* agg     = message + (size_t)N_NODES * STATE;    // 32 MB
  float* gacc    = agg + (size_t)N_NODES * STATE;        // 320 KB

  const dim3 blk(256);
  const int tileBlocks = (NTILES * 32 + 255) / 256;                  // 3907
  const int scatBlocks = (int)(((size_t)N_EDGES * 16 + 255) / 256);  // 312500
  const int gsumBlocks = (NCHUNK * 16 + 255) / 256;                  // 977
  const int outBlocks  = (N_GRAPHS + 255) / 256;                     // 20

  k_input_msg<<<tileBlocks, blk, 0, stream>>>(x, inW, inb, msgW, msgb,
                                              state, message, agg, gacc);
  for (int r = 0; r < ROUNDS; ++r) {
    k_scatter<<<scatBlocks, blk, 0, stream>>>(message, ei, agg);
    if (r < ROUNDS - 1) {
      k_upd_msg<<<tileBlocks, blk, 0, stream>>>(agg,
                                                updW + r * STATE * STATE,
                                                updb + r * STATE,
                                                msgW + (r + 1) * STATE * STATE,
                                                msgb + (r + 1) * STATE,
                                                state, message);
    } else {
      k_upd<<<tileBlocks, blk, 0, stream>>>(agg, updW + r * STATE * STATE,
                                            updb + r * STATE, state);
    }
  }
  k_gsum<<<gsumBlocks, blk, 0, stream>>>(state, batch, gacc);
  k_out<<<outBlocks, blk, 0, stream>>>(gacc, outW, outb, out);
}